// Vanilla_31095563223877
// MI455X (gfx1250) — compile-verified
//
#include <hip/hip_runtime.h>
#include <math.h>

typedef float v2f __attribute__((ext_vector_type(2)));
typedef float v8f __attribute__((ext_vector_type(8)));

#define TT 128
#define BB 256
#define VV 256
#define NN 1024

// ---------------------------------------------------------------------------
// C[M x N] = act( [A1 | A2] @ W + bias )
//   A1: M x K1 (row stride lda1), may be null when K1 == 0
//   A2: M x K2 (row stride lda2)
//   W : (K1+K2) x N, row stride N
// block = 256 threads (8 waves, 2Mx4N wave grid) computing a 32x128 C tile.
// Each wave owns 16x32 (two 16x16 accumulators sharing one A fragment).
// K staged through LDS in chunks of 32 with register double-staging so the
// next chunk's global loads overlap the current WMMA burst. W tile is held
// TRANSPOSED in LDS so each B fragment is one aligned ds_load_b64 (no VGPR
// repacking between LDS and v_wmma).
// ---------------------------------------------------------------------------
template <bool TANH>
__global__ __launch_bounds__(256) void gemm_bias_act(
    const float* __restrict__ A1, int K1, int lda1,
    const float* __restrict__ A2, int K2, int lda2,
    const float* __restrict__ W, int N,
    const float* __restrict__ bias,
    float* __restrict__ C)
{
    __shared__ float Alds[32][36];     // 32 rows x 32 K, padded
    __shared__ float WldsT[128][34];   // 128 cols x 32 K (transposed), padded

    const int tid  = threadIdx.x;
    const int lane = tid & 31;
    const int wid  = tid >> 5;
    const int mw   = wid >> 2;         // 0..1  (M sub-tile)
    const int nw   = wid & 3;          // 0..3  (N sub-tile)

    const int m0 = blockIdx.y << 5;    // 32 rows of C
    const int n0 = blockIdx.x << 7;    // 128 cols of C

    const int K = K1 + K2;

    v8f acc0 = {}, acc1 = {};

    const int halfK = (lane >= 16) ? 2 : 0;   // ISA: lanes 16-31 hold K+2,K+3
    const int rA    = (mw << 4) + (lane & 15);
    const int cB    = (nw << 5) + (lane & 15);

    float aReg[4];
    float wReg[16];

    // ---- prologue: stage chunk k0 = 0 into registers
    #pragma unroll
    for (int i = 0; i < 4; ++i) {
        int idx = tid + (i << 8);
        int r = idx >> 5, c = idx & 31;
        aReg[i] = (c < K1) ? A1[(m0 + r) * lda1 + c]
                           : A2[(m0 + r) * lda2 + (c - K1)];
    }
    #pragma unroll
    for (int i = 0; i < 16; ++i) {
        int idx = tid + (i << 8);
        int r = idx >> 7, c = idx & 127;
        wReg[i] = W[r * N + (n0 + c)];
    }

    for (int k0 = 0; k0 < K; k0 += 32) {
        // ---- commit staged registers to LDS (W stored transposed)
        #pragma unroll
        for (int i = 0; i < 4; ++i) {
            int idx = tid + (i << 8);
            Alds[idx >> 5][idx & 31] = aReg[i];
        }
        #pragma unroll
        for (int i = 0; i < 16; ++i) {
            int idx = tid + (i << 8);
            WldsT[idx & 127][idx >> 7] = wReg[i];
        }
        __syncthreads();

        // ---- prefetch next chunk into registers (uniform guard: EXEC stays
        //      all-ones; overlaps global latency with the WMMA burst below)
        const int kn = k0 + 32;
        if (kn < K) {
            #pragma unroll
            for (int i = 0; i < 4; ++i) {
                int idx = tid + (i << 8);
                int r = idx >> 5, c = idx & 31;
                int gk = kn + c;
                aReg[i] = (gk < K1) ? A1[(m0 + r) * lda1 + gk]
                                    : A2[(m0 + r) * lda2 + (gk - K1)];
            }
            #pragma unroll
            for (int i = 0; i < 16; ++i) {
                int idx = tid + (i << 8);
                int r = idx >> 7, c = idx & 127;
                wReg[i] = W[(kn + r) * N + (n0 + c)];
            }
        }

        // ---- WMMA burst over the 32-wide K chunk
        #pragma unroll
        for (int kk = 0; kk < 32; kk += 4) {
            v2f a, b0, b1;
            a.x = Alds[rA][kk + halfK];
            a.y = Alds[rA][kk + halfK + 1];
            b0  = *(const v2f*)&WldsT[cB][kk + halfK];        // ds_load_b64
            b1  = *(const v2f*)&WldsT[cB + 16][kk + halfK];   // ds_load_b64
            acc0 = __builtin_amdgcn_wmma_f32_16x16x4_f32(
                false, a, false, b0, (short)0, acc0, false, false);
            acc1 = __builtin_amdgcn_wmma_f32_16x16x4_f32(
                false, a, false, b1, (short)0, acc1, false, false);
        }
        __syncthreads();
    }

    // ---- epilogue: C/D layout -> lane<16: rows 0..7, lane>=16: rows 8..15
    const int c0 = n0 + cB;
    const int c1 = c0 + 16;
    const float bv0 = bias[c0];
    const float bv1 = bias[c1];
    const int rbase = m0 + (mw << 4) + ((lane >= 16) ? 8 : 0);
    #pragma unroll
    for (int r = 0; r < 8; ++r) {
        float v0 = acc0[r] + bv0;
        float v1 = acc1[r] + bv1;
        if (TANH) { v0 = tanhf(v0); v1 = tanhf(v1); }
        C[(rbase + r) * N + c0] = v0;
        C[(rbase + r) * N + c1] = v1;
    }
}

// ---------------------------------------------------------------------------
// Per-row log-softmax + label dot:  rowloss[row0+row] = sum_j lbl*(lse - logit)
// grid = B blocks, block = 256 threads (== V columns). Deterministic.
// ---------------------------------------------------------------------------
__global__ __launch_bounds__(256) void row_loss_kernel(
    const float* __restrict__ logits,   // B x V
    const float* __restrict__ labels,   // (T*B) x V
    float* __restrict__ rowloss,
    int row0)
{
    __shared__ float sdata[256];
    const int row = blockIdx.x;
    const int tid = threadIdx.x;

    const float l = logits[row * VV + tid];

    sdata[tid] = l;
    __syncthreads();
    for (int s = 128; s > 0; s >>= 1) {
        if (tid < s) sdata[tid] = fmaxf(sdata[tid], sdata[tid + s]);
        __syncthreads();
    }
    const float m = sdata[0];
    __syncthreads();

    sdata[tid] = __expf(l - m);
    __syncthreads();
    for (int s = 128; s > 0; s >>= 1) {
        if (tid < s) sdata[tid] += sdata[tid + s];
        __syncthreads();
    }
    const float lse = m + logf(sdata[0]);
    __syncthreads();

    const float lbl = labels[(row0 + row) * VV + tid];
    sdata[tid] = lbl * (lse - l);
    __syncthreads();
    for (int s = 128; s > 0; s >>= 1) {
        if (tid < s) sdata[tid] += sdata[tid + s];
        __syncthreads();
    }
    if (tid == 0) rowloss[row0 + row] = sdata[0];
}

__global__ __launch_bounds__(256) void final_loss_kernel(
    const float* __restrict__ rowloss, int n, float* __restrict__ out)
{
    __shared__ float sdata[256];
    const int tid = threadIdx.x;
    float p = 0.0f;
    for (int i = tid; i < n; i += 256) p += rowloss[i];
    sdata[tid] = p;
    __syncthreads();
    for (int s = 128; s > 0; s >>= 1) {
        if (tid < s) sdata[tid] += sdata[tid + s];
        __syncthreads();
    }
    if (tid == 0) out[0] = sdata[0] / (float)n;
}

__global__ void zero_kernel(float* __restrict__ p, int n)
{
    int i = blockIdx.x * blockDim.x + threadIdx.x;
    if (i < n) p[i] = 0.0f;
}

// ---------------------------------------------------------------------------
extern "C" void kernel_launch(void* const* d_in, const int* in_sizes, int n_in,
                              void* d_out, int out_size, void* d_ws, size_t ws_size,
                              hipStream_t stream)
{
    (void)in_sizes; (void)n_in; (void)out_size; (void)ws_size;

    const float* inputs  = (const float*)d_in[0];  // T x B x V
    const float* labels  = (const float*)d_in[1];  // (T*B) x V
    const float* weights = (const float*)d_in[2];  // (V+N) x N == [w_x ; w_h]
    const float* bias    = (const float*)d_in[3];  // N
    const float* w_out   = (const float*)d_in[4];  // N x V
    const float* b_out   = (const float*)d_in[5];  // V

    float* out = (float*)d_out;
    float* ws  = (float*)d_ws;

    float* h0      = ws;                       // B*N
    float* h1      = h0 + BB * NN;             // B*N
    float* logits  = h1 + BB * NN;             // B*V
    float* rowloss = logits + BB * VV;         // T*B

    zero_kernel<<<(BB * NN + 255) / 256, 256, 0, stream>>>(h0, BB * NN);

    for (int t = 0; t < TT; ++t) {
        const float* x_t = inputs + (size_t)t * BB * VV;
        float* hprev = (t & 1) ? h1 : h0;
        float* hcur  = (t & 1) ? h0 : h1;

        // h_new = tanh([x_t | h_prev] @ weights + bias)   (K = 1280)
        gemm_bias_act<true><<<dim3(NN / 128, BB / 32), 256, 0, stream>>>(
            x_t, VV, VV, hprev, NN, NN, weights, NN, bias, hcur);

        // logits = h_new @ w_out + b_out                  (K = 1024)
        gemm_bias_act<false><<<dim3(VV / 128, BB / 32), 256, 0, stream>>>(
            nullptr, 0, 0, hcur, NN, NN, w_out, VV, b_out, logits);

        row_loss_kernel<<<BB, 256, 0, stream>>>(logits, labels, rowloss, t * BB);
    }

    final_loss_kernel<<<1, 256, 0, stream>>>(rowloss, TT * BB, out);
}